// SidAgent002_26147760898721
// MI455X (gfx1250) — compile-verified
//
#include <hip/hip_runtime.h>
#include <hip/hip_bf16.h>

// ---------------------------------------------------------------------------
// LRU-style linear SSM seq2seq for MI455X (gfx1250), fp32 throughout.
// GEMMs use V_WMMA_F32_16X16X4_F32 with multi-N-tile waves (A-fragment reuse)
// and batched fragment loads so the backend can clause loads + partial-wait;
// scans exploit the complex-diagonal A (32768 independent chains).
// ---------------------------------------------------------------------------

typedef __attribute__((ext_vector_type(2))) float v2f;
typedef __attribute__((ext_vector_type(8))) float v8f;

#define T0N 512
#define T1N 512
#define NB  256
#define NY  32
#define NU  32
#define NH  256
#define NPAIR 128   // NH/2 complex channels

// ---------------------------------------------------------------------------
// fp32 WMMA GEMM:  D[m][n] = sum_k A[m][k] * B[n][k]     (B given as [N][K])
// Each wave computes one 16-row M-tile x NTW 16-col N-tiles. Per k-chunk:
// batch-load A fragment + NTW B fragments (clause-able), then NTW WMMAs.
// Epilogue: optional per-n bias, optional per-(row%NB, n) row-bias (yOff),
// optional accumulate into existing D.
// Guard is wave-uniform -> EXEC stays all-1s for WMMA (ISA 7.12 requirement).
// Fragment layouts per CDNA5 ISA 7.12.2:
//   A 16x4 f32 : lane<16 holds (M=lane, K=kk+0/kk+1), lane>=16 (M, K=kk+2/3)
//   B 4x16 f32 : lane<16 holds (K=kk+0/1, N=lane),    lane>=16 (K=kk+2/3, N)
//   C/D 16x16  : vgpr r = row (m0 + r + 8*half), col n0 + (lane&15)
// ---------------------------------------------------------------------------
template <int NTW>
__global__ void ssm_gemm_nt_wmma(const float* __restrict__ A,
                                 const float* __restrict__ B,
                                 const float* __restrict__ bias,
                                 const float* __restrict__ yOffp,
                                 float* __restrict__ D,
                                 int M, int N, int K, int accumulate)
{
    const int wavesPerBlock = blockDim.x >> 5;
    const int wave = blockIdx.x * wavesPerBlock + (threadIdx.x >> 5);
    const int lane = threadIdx.x & 31;
    const int ngroups = (N >> 4) / NTW;        // N-tile groups per M-tile
    const int mt = wave / ngroups;
    const int ng = wave - mt * ngroups;
    if (mt * 16 >= M) return;                  // wave-uniform: EXEC stays full

    const int lid  = lane & 15;
    const int half = lane >> 4;
    const int m0 = mt << 4;
    const int n0 = ng * (NTW * 16);

    v8f c[NTW];
    if (accumulate) {
        #pragma unroll
        for (int j = 0; j < NTW; ++j)
            #pragma unroll
            for (int r = 0; r < 8; ++r)
                c[j][r] = D[(size_t)(m0 + r + 8 * half) * N + n0 + j * 16 + lid];
    } else {
        #pragma unroll
        for (int j = 0; j < NTW; ++j)
            #pragma unroll
            for (int r = 0; r < 8; ++r) c[j][r] = 0.0f;
    }

    const float* __restrict__ Arow = A + (size_t)(m0 + lid) * K + 2 * half;
    const float* Brow[NTW];
    #pragma unroll
    for (int j = 0; j < NTW; ++j)
        Brow[j] = B + (size_t)(n0 + j * 16 + lid) * K + 2 * half;

    #pragma unroll 2
    for (int kk = 0; kk < K; kk += 4) {
        // Batch all fragment loads first so the backend can clause them and
        // overlap WMMA with the next chunk's loads via partial s_wait_loadcnt.
        v2f a = *(const v2f*)(Arow + kk);      // 8B aligned: K%4==0, kk%4==0
        v2f bf[NTW];
        #pragma unroll
        for (int j = 0; j < NTW; ++j)
            bf[j] = *(const v2f*)(Brow[j] + kk);
        #pragma unroll
        for (int j = 0; j < NTW; ++j)
            c[j] = __builtin_amdgcn_wmma_f32_16x16x4_f32(
                       /*neg_a=*/false, a, /*neg_b=*/false, bf[j],
                       /*c_mod=*/(short)0, c[j],
                       /*reuse_a=*/false, /*reuse_b=*/false);
    }

    #pragma unroll
    for (int j = 0; j < NTW; ++j) {
        const int n = n0 + j * 16 + lid;
        const float badd = bias ? bias[n] : 0.0f;
        #pragma unroll
        for (int r = 0; r < 8; ++r) {
            const int row = m0 + r + 8 * half;
            float v = c[j][r] + badd;
            if (yOffp) v += yOffp[(size_t)(row & (NB - 1)) * N + n];
            D[(size_t)row * N + n] = v;
        }
    }
}

// ---------------------------------------------------------------------------
// Offsets: uOff[b][u] = mean_t U0[t][b][u], yOff[b][y] = mean_t Y0[t][b][y]
// One block per batch element; 64 threads (32 U channels + 32 Y channels).
// ---------------------------------------------------------------------------
__global__ void ssm_offsets(const float* __restrict__ Y0,
                            const float* __restrict__ U0,
                            float* __restrict__ uOff,
                            float* __restrict__ yOff)
{
    const int b = blockIdx.x;
    const int t = threadIdx.x;
    float s = 0.0f;
    if (t < NU) {
        for (int k = 0; k < T0N; ++k)
            s += U0[((size_t)k * NB + b) * NU + t];
        uOff[b * NU + t] = s * (1.0f / T0N);
    } else {
        const int y = t - NU;
        for (int k = 0; k < T0N; ++k)
            s += Y0[((size_t)k * NB + b) * NY + y];
        yOff[b * NY + y] = s * (1.0f / T0N);
    }
}

// In_enc[t*NB+b][k] = k<NU ? U0c : Y0c   (offset-compensated encoder input)
__global__ void ssm_prep_enc(const float* __restrict__ U0,
                             const float* __restrict__ Y0,
                             const float* __restrict__ uOff,
                             const float* __restrict__ yOff,
                             float* __restrict__ In_enc)
{
    const size_t idx = (size_t)blockIdx.x * blockDim.x + threadIdx.x;
    const size_t total = (size_t)T0N * NB * (NU + NY);
    if (idx >= total) return;
    const int k = (int)(idx & 63);
    const size_t tb = idx >> 6;           // t*NB + b
    const int b = (int)(tb & (NB - 1));
    float v;
    if (k < NU) v = U0[tb * NU + k] - uOff[b * NU + k];
    else        v = Y0[tb * NY + (k - NU)] - yOff[b * NY + (k - NU)];
    In_enc[idx] = v;
}

// U1c = U1 - uOff
__global__ void ssm_prep_u1c(const float* __restrict__ U1,
                             const float* __restrict__ uOff,
                             float* __restrict__ U1c)
{
    const size_t idx = (size_t)blockIdx.x * blockDim.x + threadIdx.x;
    const size_t total = (size_t)T1N * NB * NU;
    if (idx >= total) return;
    const int u = (int)(idx & (NU - 1));
    const int b = (int)((idx >> 5) & (NB - 1));
    U1c[idx] = U1[idx] - uOff[b * NU + u];
}

// U2c[t][b][u], t in [0,T1]: row 0 = U0c[T0-1], rows 1.. = U1c
__global__ void ssm_prep_u2c(const float* __restrict__ U0,
                             const float* __restrict__ U1,
                             const float* __restrict__ uOff,
                             float* __restrict__ U2c)
{
    const size_t idx = (size_t)blockIdx.x * blockDim.x + threadIdx.x;
    const size_t total = (size_t)(T1N + 1) * NB * NU;
    if (idx >= total) return;
    const int u = (int)(idx & (NU - 1));
    const int b = (int)((idx >> 5) & (NB - 1));
    const int t = (int)(idx >> 13);       // /(NB*NU)
    float v = (t == 0) ? U0[((size_t)(T0N - 1) * NB + b) * NU + u]
                       : U1[((size_t)(t - 1) * NB + b) * NU + u];
    U2c[idx] = v - uOff[b * NU + u];
}

// ---------------------------------------------------------------------------
// Encoder scan: 32768 independent complex chains (b, pair p); only final state
// is needed.  x' = lambda * x + bu  (complex, diagonal)
// ---------------------------------------------------------------------------
__global__ void ssm_scan_enc(const float* __restrict__ lr_,
                             const float* __restrict__ li_,
                             const float* __restrict__ Bu,
                             float* __restrict__ xfinal)
{
    const int gid = blockIdx.x * blockDim.x + threadIdx.x;   // 0..32767
    const int b = gid >> 7;
    const int p = gid & (NPAIR - 1);
    const float lr = lr_[p], li = li_[p];
    float xr = 0.0f, xi = 0.0f;
    for (int t = 0; t < T0N; ++t) {
        const float* row = Bu + ((size_t)t * NB + b) * NH;
        const float br = row[p];
        const float bi = row[p + NPAIR];
        const float nr = fmaf(lr, xr, fmaf(-li, xi, br));
        const float ni = fmaf(li, xr, fmaf( lr, xi, bi));
        xr = nr; xi = ni;
    }
    xfinal[b * NH + p]         = xr;
    xfinal[b * NH + p + NPAIR] = xi;
}

// Decoder scan: seeded by x1init, emits every state into X2[(T1+1),NB,NH]
__global__ void ssm_scan_dec(const float* __restrict__ lr_,
                             const float* __restrict__ li_,
                             const float* __restrict__ x1init,
                             const float* __restrict__ Bu,
                             float* __restrict__ X2)
{
    const int gid = blockIdx.x * blockDim.x + threadIdx.x;
    const int b = gid >> 7;
    const int p = gid & (NPAIR - 1);
    const float lr = lr_[p], li = li_[p];
    float xr = x1init[b * NH + p];
    float xi = x1init[b * NH + p + NPAIR];
    X2[(size_t)b * NH + p]         = xr;
    X2[(size_t)b * NH + p + NPAIR] = xi;
    for (int t = 0; t < T1N; ++t) {
        const float* row = Bu + ((size_t)t * NB + b) * NH;
        const float nr = fmaf(lr, xr, fmaf(-li, xi, row[p]));
        const float ni = fmaf(li, xr, fmaf( lr, xi, row[p + NPAIR]));
        float* orow = X2 + ((size_t)(t + 1) * NB + b) * NH;
        orow[p]         = nr;
        orow[p + NPAIR] = ni;
        xr = nr; xi = ni;
    }
}

// ---------------------------------------------------------------------------
extern "C" void kernel_launch(void* const* d_in, const int* in_sizes, int n_in,
                              void* d_out, int out_size, void* d_ws, size_t ws_size,
                              hipStream_t stream)
{
    const float* Y0     = (const float*)d_in[0];
    const float* U0     = (const float*)d_in[1];
    const float* U1     = (const float*)d_in[2];
    const float* lre    = (const float*)d_in[3];
    const float* lie    = (const float*)d_in[4];
    const float* B_enc  = (const float*)d_in[5];   // [NH, NU+NY]  -> B as [N][K]
    const float* C_enc_w= (const float*)d_in[6];   // [NH, NH]
    const float* C_enc_b= (const float*)d_in[7];
    const float* lrd    = (const float*)d_in[8];
    const float* lid_   = (const float*)d_in[9];
    const float* B_dec  = (const float*)d_in[10];  // [NH, NU]
    const float* C_dec_w= (const float*)d_in[11];  // [NY, NH]
    const float* C_dec_b= (const float*)d_in[12];
    const float* D_dec_w= (const float*)d_in[13];  // [NY, NU]
    const float* D_dec_b= (const float*)d_in[14];
    float* out = (float*)d_out;

    // ---- workspace carve-up (floats) ----
    float* w = (float*)d_ws;
    size_t off = 0;
    float* uOff   = w + off; off += (size_t)NB * NU;                 // 8K
    float* yOff   = w + off; off += (size_t)NB * NY;                 // 8K
    float* In_enc = w + off; off += (size_t)T0N * NB * (NU + NY);    // 8.4M
    float* U1c    = w + off; off += (size_t)T1N * NB * NU;           // 4.2M
    float* U2c    = w + off; off += (size_t)(T1N + 1) * NB * NU;     // 4.2M
    float* Bu     = w + off; off += (size_t)T0N * NB * NH;           // 33.5M (enc, then dec)
    float* xfinal = w + off; off += (size_t)NB * NH;
    float* x1init = w + off; off += (size_t)NB * NH;
    float* X2     = w + off; off += (size_t)(T1N + 1) * NB * NH;     // 33.6M
    (void)ws_size; (void)in_sizes; (void)n_in; (void)out_size;

    const int BLK = 256;
    const int WPB = BLK / 32;   // waves per block in GEMM

    // 1) offset means
    ssm_offsets<<<NB, 64, 0, stream>>>(Y0, U0, uOff, yOff);

    // 2) encoder input (offset-compensated concat)
    {
        size_t total = (size_t)T0N * NB * (NU + NY);
        ssm_prep_enc<<<(unsigned)((total + BLK - 1) / BLK), BLK, 0, stream>>>(
            U0, Y0, uOff, yOff, In_enc);
    }

    // 3) Bu_enc = In_enc @ B_enc^T : M=131072, N=256, K=64   (NTW=4)
    {
        int M = T0N * NB, N = NH, K = NU + NY;
        int waves = (M / 16) * ((N / 16) / 4);
        ssm_gemm_nt_wmma<4><<<(waves + WPB - 1) / WPB, BLK, 0, stream>>>(
            In_enc, B_enc, nullptr, nullptr, Bu, M, N, K, 0);
    }

    // 4) encoder scan -> final state only
    ssm_scan_enc<<<(NB * NPAIR) / BLK, BLK, 0, stream>>>(lre, lie, Bu, xfinal);

    // 5) x1init = xfinal @ C_enc_w^T + C_enc_b : M=256, N=256, K=256  (NTW=4)
    {
        int M = NB, N = NH, K = NH;
        int waves = (M / 16) * ((N / 16) / 4);
        ssm_gemm_nt_wmma<4><<<(waves + WPB - 1) / WPB, BLK, 0, stream>>>(
            xfinal, C_enc_w, C_enc_b, nullptr, x1init, M, N, K, 0);
    }

    // 6) decoder input
    {
        size_t total = (size_t)T1N * NB * NU;
        ssm_prep_u1c<<<(unsigned)((total + BLK - 1) / BLK), BLK, 0, stream>>>(
            U1, uOff, U1c);
    }

    // 7) Bu_dec = U1c @ B_dec^T : M=131072, N=256, K=32   (NTW=4, reuses Bu)
    {
        int M = T1N * NB, N = NH, K = NU;
        int waves = (M / 16) * ((N / 16) / 4);
        ssm_gemm_nt_wmma<4><<<(waves + WPB - 1) / WPB, BLK, 0, stream>>>(
            U1c, B_dec, nullptr, nullptr, Bu, M, N, K, 0);
    }

    // 8) decoder scan -> full X2 trajectory
    ssm_scan_dec<<<(NB * NPAIR) / BLK, BLK, 0, stream>>>(lrd, lid_, x1init, Bu, X2);

    // 9) feed-through input U2c
    {
        size_t total = (size_t)(T1N + 1) * NB * NU;
        ssm_prep_u2c<<<(unsigned)((total + BLK - 1) / BLK), BLK, 0, stream>>>(
            U0, U1, uOff, U2c);
    }

    // 10) out = X2 @ C_dec_w^T + C_dec_b : M=131328, N=32, K=256   (NTW=2)
    {
        int M = (T1N + 1) * NB, N = NY, K = NH;
        int waves = (M / 16) * ((N / 16) / 2);
        ssm_gemm_nt_wmma<2><<<(waves + WPB - 1) / WPB, BLK, 0, stream>>>(
            X2, C_dec_w, C_dec_b, nullptr, out, M, N, K, 0);
    }

    // 11) out += U2c @ D_dec_w^T + D_dec_b[y] + yOff[b][y]  (NTW=2, accumulate;
    //     bias + per-row offset folded into the store epilogue -> no extra pass)
    {
        int M = (T1N + 1) * NB, N = NY, K = NU;
        int waves = (M / 16) * ((N / 16) / 2);
        ssm_gemm_nt_wmma<2><<<(waves + WPB - 1) / WPB, BLK, 0, stream>>>(
            U2c, D_dec_w, D_dec_b, yOff, out, M, N, K, 1);
    }
}